// privacyLoss1_8203387535487
// MI455X (gfx1250) — compile-verified
//
#include <hip/hip_runtime.h>
#include <hip/hip_bf16.h>

typedef __attribute__((ext_vector_type(2))) float v2f;
typedef __attribute__((ext_vector_type(8))) float v8f;

#define B_ROWS 4096
#define D_DIM  256
#define NLAB   16          // pad 10 labels to 16 for the WMMA tile
#define RPB    64          // rows per block in the accumulation pass

// workspace layout (floats)
#define WS_SALL  0                       // [16][256] per-label feature sums (all rows)
#define WS_SQCOL (NLAB * D_DIM)          // [16][256] per-label per-column sum of squares
#define WS_CNT   (2 * NLAB * D_DIM)      // [16] per-label counts (as float)
#define WS_TOTAL (2 * NLAB * D_DIM + NLAB)

__global__ void pl_zero_ws(float* ws, int n) {
    int i = blockIdx.x * blockDim.x + threadIdx.x;
    if (i < n) ws[i] = 0.0f;
}

// One streaming pass over feature[4096][256]. Thread d owns column d of every
// per-label LDS accumulator, so no intra-block races; rows are walked
// sequentially per block. Global fp32 atomics merge blocks.
__global__ void pl_accum(const float* __restrict__ feat,
                         const int* __restrict__ lab,
                         float* __restrict__ ws) {
    __shared__ float sAll[NLAB][D_DIM];   // 16 KB
    __shared__ float sqp [NLAB][D_DIM];   // 16 KB
    __shared__ int   cnt [NLAB];

    const int d = threadIdx.x;            // 256 threads
    for (int c = 0; c < NLAB; ++c) { sAll[c][d] = 0.0f; sqp[c][d] = 0.0f; }
    if (d < NLAB) cnt[d] = 0;
    __syncthreads();

    const int base = blockIdx.x * RPB;
    for (int r = 0; r < RPB; ++r) {
        const int i = base + r;
        const int c = lab[i] & (NLAB - 1);
        const float v = feat[i * D_DIM + d];    // coalesced 1 KB/row
        sAll[c][d] += v;
        sqp [c][d] += v * v;
        if (d == 0) cnt[c] += 1;                // only thread 0 touches cnt
    }
    __syncthreads();

    for (int c = 0; c < NLAB; ++c) {
        atomicAdd(&ws[WS_SALL  + c * D_DIM + d], sAll[c][d]);
        atomicAdd(&ws[WS_SQCOL + c * D_DIM + d], sqp [c][d]);
    }
    if (d < NLAB) atomicAdd(&ws[WS_CNT + d], (float)cnt[d]);
}

// Single wave (EXEC all-ones, as WMMA requires). Builds per-label sums for the
// i-range (drop last row) and j-range (drop first row), computes the 16x16
// label Gram G[a][b] = sI(a).sJ(b) with V_WMMA_F32_16X16X4_F32 over K=256,
// then assembles the scalar loss.
__global__ void pl_finalize(const float* __restrict__ feat,
                            const int* __restrict__ lab,
                            const float* __restrict__ ws,
                            float* __restrict__ out) {
    __shared__ float sI[NLAB][D_DIM];     // 16 KB
    __shared__ float sJ[NLAB][D_DIM];     // 16 KB
    __shared__ float G[NLAB][NLAB];       // 1 KB

    const int lane = threadIdx.x;         // 32 threads
    const int l0 = lab[0]          & (NLAB - 1);
    const int lB = lab[B_ROWS - 1] & (NLAB - 1);

    // sI(c) = sAll(c) - [c==lB]*f_last ; sJ(c) = sAll(c) - [c==l0]*f_first
    for (int c = 0; c < NLAB; ++c) {
        for (int d = lane; d < D_DIM; d += 32) {
            const float s = ws[WS_SALL + c * D_DIM + d];
            const float cI = (c == lB) ? feat[(B_ROWS - 1) * D_DIM + d] : 0.0f;
            const float cJ = (c == l0) ? feat[d] : 0.0f;
            sI[c][d] = s - cI;
            sJ[c][d] = s - cJ;
        }
    }
    __syncthreads();

    // 16x16x256 fp32 Gram via 64 accumulating WMMAs.
    // A (16x4): lanes 0-15 -> row=lane, K=k0..k0+1 ; lanes 16-31 -> row=lane-16, K=k0+2..k0+3.
    // B (4x16): identical per-lane index pattern, sourced from sJ.
    v8f acc = {};
    const int row  = lane & 15;
    const int koff = (lane >> 4) * 2;
    for (int k0 = 0; k0 < D_DIM; k0 += 4) {
        v2f a, b;
        a.x = sI[row][k0 + koff];
        a.y = sI[row][k0 + koff + 1];
        b.x = sJ[row][k0 + koff];
        b.y = sJ[row][k0 + koff + 1];
        acc = __builtin_amdgcn_wmma_f32_16x16x4_f32(
            /*neg_a=*/false, a, /*neg_b=*/false, b,
            /*c_mod=*/(short)0, acc, /*reuse_a=*/false, /*reuse_b=*/false);
    }

    // Spill D: VGPR v, lanes 0-15 -> M=v ; lanes 16-31 -> M=v+8 ; N=lane&15.
    const int m0 = (lane < 16) ? 0 : 8;
    const int n  = lane & 15;
#pragma unroll
    for (int v = 0; v < 8; ++v) G[m0 + v][n] = acc[v];
    __syncthreads();

    if (lane == 0) {
        // squared norms of first/last rows for the sq corrections
        float sqf0 = 0.0f, sqfB = 0.0f;
        for (int d = 0; d < D_DIM; ++d) {
            const float a0 = feat[d];
            const float aB = feat[(B_ROWS - 1) * D_DIM + d];
            sqf0 += a0 * a0;
            sqfB += aB * aB;
        }
        float SqI_tot = 0.0f, SqJ_tot = 0.0f, S_same = 0.0f, totDot = 0.0f;
        for (int c = 0; c < NLAB; ++c) {
            float sq = 0.0f;
            for (int d = 0; d < D_DIM; ++d) sq += ws[WS_SQCOL + c * D_DIM + d];
            const float nAll = ws[WS_CNT + c];
            const float sqI = sq - ((c == lB) ? sqfB : 0.0f);
            const float sqJ = sq - ((c == l0) ? sqf0 : 0.0f);
            const float nI  = nAll - ((c == lB) ? 1.0f : 0.0f);
            const float nJ  = nAll - ((c == l0) ? 1.0f : 0.0f);
            SqI_tot += sqI;
            SqJ_tot += sqJ;
            S_same  += nJ * sqI + nI * sqJ - 2.0f * G[c][c];
        }
        for (int a = 0; a < NLAB; ++a)
            for (int b = 0; b < NLAB; ++b) totDot += G[a][b];

        const float nRange = (float)(B_ROWS - 1);   // |I| = |J| = 4095
        const float S_all  = nRange * SqI_tot + nRange * SqJ_tot - 2.0f * totDot;
        const float par    = 0.5f / (float)B_ROWS;  // SIGMA * 0.5 / batch
        out[0] = par * (S_all - 2.0f * S_same);
    }
}

extern "C" void kernel_launch(void* const* d_in, const int* in_sizes, int n_in,
                              void* d_out, int out_size, void* d_ws, size_t ws_size,
                              hipStream_t stream) {
    const float* feat = (const float*)d_in[0];
    const int*   lab  = (const int*)d_in[1];
    float*       out  = (float*)d_out;
    float*       ws   = (float*)d_ws;

    pl_zero_ws<<<(WS_TOTAL + 255) / 256, 256, 0, stream>>>(ws, WS_TOTAL);
    pl_accum  <<<B_ROWS / RPB, D_DIM, 0, stream>>>(feat, lab, ws);
    pl_finalize<<<1, 32, 0, stream>>>(feat, lab, ws, out);
}